// MimickedSelfContactLoss_88931592831350
// MI455X (gfx1250) — compile-verified
//
#include <hip/hip_runtime.h>
#include <hip/hip_bf16.h>
#include <math.h>

#define NV 10475
#define NC 1024
#define ROW_TILES (NC / 16)          // 64
#define N_TILES ((NV + 15) / 16)     // 655
#define WPB 8                        // wave32s per block
#define GRID_Y 11                    // 64x11 blocks -> 88 wave slots over 655 tiles

typedef __attribute__((ext_vector_type(2))) float v2f;
typedef __attribute__((ext_vector_type(8))) float v8f;

// Explicit global-address-space pointer: guarantees mask gathers lower as
// global_load_b32 (LOADcnt only), never flat_load (which also ties up DScnt).
typedef __attribute__((address_space(1))) const int gmem_int;

// ---------------------------------------------------------------------------
// Kernel 0: initialize per-contact-row packed (d2_bits << 32 | j) keys to MAX.
// ---------------------------------------------------------------------------
__global__ void init_ws_kernel(unsigned long long* __restrict__ keys) {
    int t = blockIdx.x * blockDim.x + threadIdx.x;
    if (t < NC) keys[t] = 0xFFFFFFFFFFFFFFFFull;
}

// ---------------------------------------------------------------------------
// Kernel 1: WMMA-based masked nearest-neighbor selection.
//   16x16 dot tile via V_WMMA_F32_16X16X4_F32 (K=3 real + 1 zero pad),
//   d2 = |a|^2 + |b|^2 - 2*dot, branchless masked running min per row,
//   packed u64 atomicMin for the global combine (monotone for d2 >= 0;
//   low bits = column index j -> ties resolve to smallest j, matching
//   jnp.argmin first-occurrence).
// ---------------------------------------------------------------------------
__global__ __launch_bounds__(256)
void contact_min_kernel(const int* __restrict__ contacts,
                        const float* __restrict__ verts,   // [NV][3]
                        const int* __restrict__ gmask,     // [NV][NV] 0/1
                        unsigned long long* __restrict__ keys) {
    __shared__ int   s_ci[16];   // contact vertex ids of this row tile
    __shared__ float s_ra[16];   // |a|^2 per row

    const int m0  = blockIdx.x * 16;
    const int tid = threadIdx.x;

    if (tid < 16) {
        int ci = contacts[m0 + tid];
        s_ci[tid] = ci;
        float x = verts[3 * ci], y = verts[3 * ci + 1], z = verts[3 * ci + 2];
        s_ra[tid] = x * x + y * y + z * z;
    }
    __syncthreads();

    const int wave  = tid >> 5;
    const int ln    = tid & 31;
    const int half  = ln >> 4;      // 0: K=0,1 / rows 0..7 ; 1: K=2,3 / rows 8..15
    const int nlane = ln & 15;
    const int baseM = half * 8;

    // A fragment: 16x4 f32. lanes 0-15 -> (K0,K1), lanes 16-31 -> (K2,K3=0)
    const int ciA = s_ci[nlane];
    const float ax = verts[3 * ciA], ay = verts[3 * ciA + 1], az = verts[3 * ciA + 2];
    v2f A;
    A.x = half ? az : ax;
    A.y = half ? 0.0f : ay;

    // Loop-invariant per-row state in registers (no LDS re-loads in the loop):
    //   ra[r]   = |a|^2 of the row this lane's D element r belongs to
    //   rowg[r] = that row's geomask base as an explicit AS(1) pointer
    float     ra[8];
    gmem_int* rowg[8];
#pragma unroll
    for (int r = 0; r < 8; ++r) {
        const int row = baseM + r;
        ra[r]   = s_ra[row];
        rowg[r] = (gmem_int*)(unsigned long long)(gmask + (long long)s_ci[row] * NV);
    }

    float minv[8];
    int   minj[8];
#pragma unroll
    for (int r = 0; r < 8; ++r) { minv[r] = __builtin_inff(); minj[r] = 0; }

    const float INF = __builtin_inff();

    for (int t = blockIdx.y * WPB + wave; t < N_TILES; t += GRID_Y * WPB) {
        const int  n     = t * 16 + nlane;
        const bool valid = (n < NV);
        const int  nc    = valid ? n : (NV - 1);   // clamped: loads stay in
                                                   // bounds; invalid lanes are
                                                   // neutralized arithmetically.

        const float bx = verts[3 * nc];
        const float by = verts[3 * nc + 1];
        const float bz = verts[3 * nc + 2];

        // B fragment: 4x16 f32, mirrored layout of A.
        v2f B;
        B.x = half ? bz : bx;
        B.y = half ? 0.0f : by;
        const float rb = bx * bx + by * by + bz * bz;

        v8f acc = {0.f, 0.f, 0.f, 0.f, 0.f, 0.f, 0.f, 0.f};
        acc = __builtin_amdgcn_wmma_f32_16x16x4_f32(
            /*neg_a=*/false, A, /*neg_b=*/false, B,
            /*c_mod=*/(short)0, acc, /*reuse_a=*/false, /*reuse_b=*/false);

        // Batch-issue 8 independent global NT gathers (staggered loadcnt waits).
        int mg[8];
#pragma unroll
        for (int r = 0; r < 8; ++r)
            mg[r] = __builtin_nontemporal_load(rowg[r] + nc);

        // Branchless masked min update (no exec-mask manipulation).
#pragma unroll
        for (int r = 0; r < 8; ++r) {
            const float d2   = fmaxf(ra[r] + rb - 2.0f * acc[r], 0.0f);
            const float cand = (valid && (mg[r] != 0)) ? d2 : INF;
            const bool  take = cand < minv[r];   // n strictly increases per lane,
            minv[r] = take ? cand : minv[r];     // so '<' keeps the earliest j.
            minj[r] = take ? n : minj[r];
        }
    }

    // Cross-lane min over the 16 lanes sharing each row (XOR stays within half).
#pragma unroll
    for (int off = 8; off >= 1; off >>= 1) {
#pragma unroll
        for (int r = 0; r < 8; ++r) {
            const float ov = __shfl_xor(minv[r], off, 32);
            const int   oj = __shfl_xor(minj[r], off, 32);
            const bool take = (ov < minv[r]) || (ov == minv[r] && oj < minj[r]);
            minv[r] = take ? ov : minv[r];
            minj[r] = take ? oj : minj[r];
        }
    }

    if (nlane == 0) {  // lanes 0 and 16 publish rows 0..7 / 8..15
#pragma unroll
        for (int r = 0; r < 8; ++r) {
            const unsigned long long key =
                ((unsigned long long)__float_as_uint(minv[r]) << 32) |
                (unsigned long long)(unsigned int)minj[r];
            atomicMin(&keys[m0 + baseM + r], key);
        }
    }
}

// ---------------------------------------------------------------------------
// Kernel 2: exact recompute of the winning distance (matches _safe_norm
// bit-path: sqrt of sum of squared diffs), tanh, mean over 1024 contacts.
// ---------------------------------------------------------------------------
__global__ __launch_bounds__(1024)
void finalize_kernel(const int* __restrict__ contacts,
                     const float* __restrict__ verts,
                     const unsigned long long* __restrict__ keys,
                     float* __restrict__ out) {
    __shared__ float s_part[32];
    const int t = threadIdx.x;

    const unsigned long long key = keys[t];
    const unsigned int fbits = (unsigned int)(key >> 32);
    int j = (int)(key & 0xFFFFFFFFu);
    if (fbits >= 0x7F800000u) j = 0;  // empty mask row: argmin of all-inf == 0

    const int ci = contacts[t];
    const float dx = verts[3 * ci]     - verts[3 * j];
    const float dy = verts[3 * ci + 1] - verts[3 * j + 1];
    const float dz = verts[3 * ci + 2] - verts[3 * j + 2];
    const float sq = dx * dx + dy * dy + dz * dz;
    const float d  = (sq > 0.0f) ? sqrtf(sq) : 0.0f;
    float val = tanhf(d);  // CONTACT_THRESH == 1.0

#pragma unroll
    for (int off = 16; off >= 1; off >>= 1) val += __shfl_down(val, off, 32);
    if ((t & 31) == 0) s_part[t >> 5] = val;
    __syncthreads();
    if (t < 32) {
        float v = s_part[t];
#pragma unroll
        for (int off = 16; off >= 1; off >>= 1) v += __shfl_down(v, off, 32);
        if (t == 0) out[0] = v * (1.0f / (float)NC);
    }
}

// ---------------------------------------------------------------------------
extern "C" void kernel_launch(void* const* d_in, const int* in_sizes, int n_in,
                              void* d_out, int out_size, void* d_ws, size_t ws_size,
                              hipStream_t stream) {
    (void)in_sizes; (void)n_in; (void)out_size; (void)ws_size;

    const int*   contacts = (const int*)d_in[0];    // presented_contact [1024]
    const float* verts    = (const float*)d_in[1];  // vertices [1,10475,3]
    const int*   gmask    = (const int*)d_in[2];    // geomask [10475,10475]
    float*       out      = (float*)d_out;

    unsigned long long* keys = (unsigned long long*)d_ws;  // 1024 * u64 = 8KB

    init_ws_kernel<<<1, 1024, 0, stream>>>(keys);

    dim3 grid(ROW_TILES, GRID_Y);
    contact_min_kernel<<<grid, 256, 0, stream>>>(contacts, verts, gmask, keys);

    finalize_kernel<<<1, 1024, 0, stream>>>(contacts, verts, keys, out);
}